// CRNNDecoder_25417616458110
// MI455X (gfx1250) — compile-verified
//
#include <hip/hip_runtime.h>
#include <hip/hip_bf16.h>
#include <stdint.h>

// CTC greedy decode: logits [B, C=12, T=2048] f32 -> decoded [B,6] + entropy-confidences [B,6].
// One wave32 per batch row. Per 32-timestep tile:
//   - async-copy (global_load_async_to_lds_b128) the 12x32 f32 class tile into a
//     per-wave LDS slice; issued on demand (rows almost always finish in tile 0,
//     so speculative prefetch would only double HBM traffic)
//   - per-lane argmax over 12 classes + softmax entropy
//   - dedup/blank mask -> ballot_w32 + popc prefix scan for compaction position
//   - early-exit the row as soon as 6 tokens are kept (reference drops the rest)

#define kC        12
#define kT        2048
#define kMAXLEN   6
#define kBLANK    11
#define kTILE     32            // timesteps per tile (one per lane)
#define kWAVES    8             // waves (rows) per block
#define kNTILES   (kT / kTILE)  // 64

// Issue 3 async b128 loads covering one 12 x 32 f32 tile (1536 B):
// instruction k: lanes -> class 4k + lane/8, 16B quad (lane%8) of that class row.
__device__ __forceinline__ void issue_tile_async(const float* __restrict__ row,
                                                 unsigned ldsBase, int tile, int lane) {
    const int q    = lane & 7;   // 16-byte quad within the 128B class row
    const int csub = lane >> 3;  // 0..3
#pragma unroll
    for (int k = 0; k < 3; ++k) {
        const int c = 4 * k + csub;
        const float* g = row + (size_t)c * kT + (size_t)tile * kTILE + q * 4;
        unsigned long long gaddr = (unsigned long long)(uintptr_t)g;
        unsigned lds = ldsBase + (unsigned)(c * (kTILE * 4) + q * 16);
        asm volatile("global_load_async_to_lds_b128 %0, %1, off"
                     :: "v"(lds), "v"(gaddr) : "memory");
    }
}

__global__ __launch_bounds__(kWAVES * 32)
void ctc_greedy_decode_kernel(const float* __restrict__ logits,
                              float* __restrict__ out, int B) {
    __shared__ float lbuf[kWAVES][kC * kTILE];  // 8 * 1536B = 12 KB

    const int lane = threadIdx.x & 31;
    const int wave = threadIdx.x >> 5;
    const int b    = blockIdx.x * kWAVES + wave;
    if (b >= B) return;  // wave-uniform

    float* outDec  = out;                          // decoded  [B,6] (as float)
    float* outConf = out + (size_t)B * kMAXLEN;    // entropy  [B,6]

    // defaults: PAD_IDX = -1, confidence = 0 (harness poisons d_out)
    if (lane < kMAXLEN) {
        outDec [b * kMAXLEN + lane] = -1.0f;
        outConf[b * kMAXLEN + lane] =  0.0f;
    }

    const float* row = logits + (size_t)b * kC * kT;
    // low 32 bits of a flat pointer into LDS are the raw LDS byte offset
    const unsigned ldsB = (unsigned)(uintptr_t)&lbuf[wave][0];
    const float* buf = &lbuf[wave][0];

    issue_tile_async(row, ldsB, 0, lane);

    int prevCarry = -1;  // pred of last timestep of previous tile (-1 sentinel)
    int count     = 0;   // tokens kept so far in this row

    for (int tile = 0; tile < kNTILES; ++tile) {
        asm volatile("s_wait_asynccnt 0x0" ::: "memory");  // tile landed in LDS

        // argmax over classes (strict > => first max, matches jnp.argmax)
        float v[kC];
        float m   = -3.402823466e38f;
        int   arg = 0;
#pragma unroll
        for (int c = 0; c < kC; ++c) {
            float x = buf[c * kTILE + lane];
            v[c] = x;
            if (x > m) { m = x; arg = c; }
        }

        // entropy of softmax: -sum p * log(p + 1e-6)
        float s = 0.0f;
#pragma unroll
        for (int c = 0; c < kC; ++c) { v[c] = __expf(v[c] - m); s += v[c]; }
        const float inv = __builtin_amdgcn_rcpf(s);
        float H = 0.0f;
#pragma unroll
        for (int c = 0; c < kC; ++c) {
            float p = v[c] * inv;
            H -= p * __logf(p + 1e-6f);
        }

        // dedup: keep first of each run, drop blanks
        int pprev = __shfl_up(arg, 1);
        if (lane == 0) pprev = prevCarry;
        const bool keep = (arg != pprev) && (arg != kBLANK);

        unsigned ball = __builtin_amdgcn_ballot_w32(keep);
        int pos = count + __popc(ball & ((1u << lane) - 1u));
        if (keep && pos < kMAXLEN) {
            outDec [b * kMAXLEN + pos] = (float)arg;
            outConf[b * kMAXLEN + pos] = H;
        }
        count += __popc(ball);
        if (count >= kMAXLEN) break;  // row done; nothing in flight at this point
        prevCarry = __shfl(arg, 31);

        // only fetch the next tile if this row actually needs it
        // (all 12 DS-load results above are already consumed, so the LDS
        //  buffer reuse has no WAR hazard against the new async writes)
        if (tile + 1 < kNTILES) issue_tile_async(row, ldsB, tile + 1, lane);
    }
}

extern "C" void kernel_launch(void* const* d_in, const int* in_sizes, int n_in,
                              void* d_out, int out_size, void* d_ws, size_t ws_size,
                              hipStream_t stream) {
    const float* logits = (const float*)d_in[0];
    float* out = (float*)d_out;
    const int B = in_sizes[0] / (kC * kT);   // 2048
    const int blocks = (B + kWAVES - 1) / kWAVES;
    hipLaunchKernelGGL(ctc_greedy_decode_kernel, dim3(blocks), dim3(kWAVES * 32),
                       0, stream, logits, out, B);
}